// ContrastiveCRFLoss_23381801959612
// MI455X (gfx1250) — compile-verified
//
#include <hip/hip_runtime.h>
#include <hip/hip_bf16.h>

// ---------------------------------------------------------------------------
// ContrastiveCRFLoss on MI455X (gfx1250)
//
// out[n,a,b] = -( (Σ_k selC[n,k,a]·selC[n,k,b]) · w(a,b,n) )
//   w = 10·exp(-cd - gd/0.3) + 3·exp(-cd/50),  cd = coord dist², gd = guidance dist²
//
// Roofline: 134 MB output @ 23.3 TB/s => ~5.8 us floor; GEMM is only 1.8 GFLOP
// (K=27), done with V_WMMA_F32_16X16X4_F32 (f32-in/f32-acc, full precision).
// cd is batch-independent, so each wave computes one 16x16 (a,b) tile for all
// 8 batches and reuses the two coordinate exps 8x (cuts v_exp count ~40%).
// Gathered operands (~2 MB) are L2-resident; stores are non-temporal.
// ---------------------------------------------------------------------------

typedef float v2f __attribute__((ext_vector_type(2)));
typedef float v8f __attribute__((ext_vector_type(8)));

#define NS   2048     // number of samples
#define KC   27       // cluster channels
#define KPAD 28       // padded K (multiple of 4 for 16x16x4 WMMA chunks)

// workspace layout (floats):
//   fx   : [0,        NS)                coords row 0 as float
//   fy   : [NS,       2*NS)              coords row 1 as float
//   g    : [2*NS,     2*NS + 24*NS)      g[n][ch][s], n<8, ch<3
//   selC : [26*NS,    26*NS + 8*NS*28)   selC[n][s][k], k<28 (k=27 zeroed)

__global__ void __launch_bounds__(256)
crf_gather_kernel(const float* __restrict__ guidance,
                  const float* __restrict__ clusters,
                  const int*   __restrict__ coords,
                  float*       __restrict__ ws) {
    int tid = blockIdx.x * blockDim.x + threadIdx.x;   // 0..16383
    int n = tid >> 11;                                  // batch 0..7
    int s = tid & (NS - 1);                             // sample 0..2047

    int x = coords[s];
    int y = coords[NS + s];

    float* fx   = ws;
    float* fy   = ws + NS;
    float* g    = ws + 2 * NS;
    float* selC = ws + 26 * NS;

    if (n == 0) { fx[s] = (float)x; fy[s] = (float)y; }

    long pix = (long)x * 224 + y;

    // guidance gather: g[n][ch][s]
    #pragma unroll
    for (int ch = 0; ch < 3; ++ch)
        g[(n * 3 + ch) * NS + s] =
            guidance[((long)(n * 3 + ch)) * (224 * 224) + pix];

    // cluster gather into K-padded row-per-sample layout: selC[n][s][k]
    float* row = selC + (size_t)(n * NS + s) * KPAD;
    #pragma unroll
    for (int k = 0; k < KC; ++k)
        row[k] = clusters[((long)(n * KC + k)) * (224 * 224) + pix];
    row[KC] = 0.0f;   // zero-pad k=27
}

__global__ void __launch_bounds__(256)
crf_tile_kernel(const float* __restrict__ ws,
                float*       __restrict__ out) {
    const int lane = threadIdx.x & 31;
    const int wave = threadIdx.x >> 5;

    // one wave per 16x16 (a,b) tile, covering all 8 batches
    int unit = blockIdx.x * 8 + wave;        // 0..16383
    int ta = unit >> 7;                      // 0..127
    int tb = unit & 127;                     // 0..127
    int a0 = ta << 4;
    int b0 = tb << 4;

    const float* fx   = ws;
    const float* fy   = ws + NS;
    const float* g    = ws + 2 * NS;
    const float* selC = ws + 26 * NS;

    const int m  = lane & 15;   // within-tile column (b) / fragment row (a)
    const int hi = lane >> 4;   // half-wave select
    const int kb = hi * 2;      // K sub-offset for 16x16x4 f32 A/B layout

    // ---- batch-independent coordinate kernel (computed once, reused 8x) ----
    const int bi = b0 + m;
    const float bx = fx[bi], by = fy[bi];
    float e1[8], e2[8];
    #pragma unroll
    for (int v = 0; v < 8; ++v) {
        int ai = a0 + v + 8 * hi;            // C/D layout: VGPR v -> row M
        float dx = fx[ai] - bx;
        float dy = fy[ai] - by;
        float cd = dx * dx + dy * dy;
        e1[v] = 10.0f * __expf(-cd);         // W1 * exp(-cd/(2*ALPHA)), 2A=1.0
        e2[v] =  3.0f * __expf(-cd * 0.02f); // W2 * exp(-cd/(2*GAMMA)), 2G=50
    }

    for (int n = 0; n < 8; ++n) {
        // ---- Gram tile via 7x V_WMMA_F32_16X16X4_F32 (K = 28 padded) ----
        const float* Arow = selC + (size_t)(n * NS + a0 + m) * KPAD + kb;
        const float* Brow = selC + (size_t)(n * NS + b0 + m) * KPAD + kb;

        v8f acc = {0.f, 0.f, 0.f, 0.f, 0.f, 0.f, 0.f, 0.f};
        #pragma unroll
        for (int c = 0; c < 7; ++c) {
            v2f af = *(const v2f*)(Arow + c * 4);
            v2f bf = *(const v2f*)(Brow + c * 4);
            acc = __builtin_amdgcn_wmma_f32_16x16x4_f32(
                /*neg_a=*/false, af, /*neg_b=*/false, bf,
                /*c_mod=*/(short)0, acc, /*reuse_a=*/false, /*reuse_b=*/false);
        }

        // ---- guidance kernel + fused scale/negate + streamed store ----
        const float* gn = g + n * 3 * NS;
        float bg0 = gn[bi];
        float bg1 = gn[NS + bi];
        float bg2 = gn[2 * NS + bi];

        float* obase = out + ((size_t)n * NS) * NS + bi;
        #pragma unroll
        for (int v = 0; v < 8; ++v) {
            int ai = a0 + v + 8 * hi;
            float d0 = gn[ai]          - bg0;
            float d1 = gn[NS + ai]     - bg1;
            float d2 = gn[2 * NS + ai] - bg2;
            float gd = d0 * d0 + d1 * d1 + d2 * d2;
            // w = e1*exp(-gd/(2*BETA)) + e2, 2*BETA = 0.3
            float w = e1[v] * __expf(-gd * (1.0f / 0.3f)) + e2[v];
            __builtin_nontemporal_store(-(acc[v] * w), obase + (size_t)ai * NS);
        }
    }
}

extern "C" void kernel_launch(void* const* d_in, const int* in_sizes, int n_in,
                              void* d_out, int out_size, void* d_ws, size_t ws_size,
                              hipStream_t stream) {
    const float* guidance = (const float*)d_in[0];   // (8,3,224,224) f32
    const float* clusters = (const float*)d_in[1];   // (8,27,224,224) f32
    const int*   coords   = (const int*)d_in[2];     // (2,2048) int
    float* out = (float*)d_out;                      // (8,2048,2048) f32
    float* ws  = (float*)d_ws;                       // ~2.1 MB used

    // gather: 8 batches x 2048 samples, one thread each
    crf_gather_kernel<<<64, 256, 0, stream>>>(guidance, clusters, coords, ws);

    // tiles: 128x128 16x16 tiles, 8 waves (256 thr) per block
    crf_tile_kernel<<<2048, 256, 0, stream>>>(ws, out);
}